// CUDANTMCell_54906861912661
// MI455X (gfx1250) — compile-verified
//
#include <hip/hip_runtime.h>
#include <math.h>

typedef float v2f __attribute__((ext_vector_type(2)));
typedef float v8f __attribute__((ext_vector_type(8)));

#define T_DIM 1024
#define B_DIM 64
#define D_DIM 512
#define N_DIM 64
#define ROWS  (T_DIM * B_DIM)          // 65536 rows of the projection GEMM
#define TBN   ((size_t)T_DIM * B_DIM * N_DIM)  // 4194304 elems per activation

// ---------------------------------------------------------------------------
// Kernel 1: fused 5-way projection GEMM (fp32 WMMA 16x16x4) + activations.
// grid.x = ROWS/16 workgroups; block = 128 threads = 4 waves (wave32).
// Wave w computes the 16(M) x 16(N) tile at Nbase = 16*w for ALL FIVE weight
// matrices (A-fragment reuse), accumulating over K = 512 in steps of 4.
// ---------------------------------------------------------------------------
__global__ __launch_bounds__(128) void ntm_proj_kernel(
    const float* __restrict__ x,
    const float* __restrict__ Wk, const float* __restrict__ Wv,
    const float* __restrict__ Wq, const float* __restrict__ We,
    const float* __restrict__ Ww,
    float* __restrict__ outk, float* __restrict__ outv,
    float* __restrict__ outq, float* __restrict__ oute,
    float* __restrict__ outw)
{
    __shared__ float ldsA[16 * 64];        // x chunk: 16 rows x 64 K-columns
    __shared__ float ldsS[2 * 16 * 64];    // softmax staging for k and q

    const int tid   = threadIdx.x;
    const int wave  = tid >> 5;
    const int lane  = tid & 31;
    const int lmod  = lane & 15;           // row (A) / column (B,C,D) index 0..15
    const bool hi   = lane >= 16;
    const int Mbase = blockIdx.x * 16;
    const int Nbase = wave * 16;

    const float* Wmat[5] = {Wk, Wv, Wq, We, Ww};

    v8f acc[5];
#pragma unroll
    for (int m = 0; m < 5; ++m)
#pragma unroll
        for (int r = 0; r < 8; ++r) acc[m][r] = 0.0f;

    // K loop: 8 chunks of 64, staged through LDS
    for (int c = 0; c < 8; ++c) {
        __syncthreads();
#pragma unroll
        for (int i = tid; i < 16 * 64; i += 128) {
            const int r = i >> 6, col = i & 63;
            ldsA[i] = x[(size_t)(Mbase + r) * D_DIM + c * 64 + col];
        }
        __syncthreads();

        for (int kk = 0; kk < 64; kk += 4) {
            // A fragment: lanes 0-15 hold K=kk,kk+1 of row lmod; lanes 16-31 hold K=kk+2,kk+3
            const int ak = kk + (hi ? 2 : 0);
            v2f a;
            a.x = ldsA[lmod * 64 + ak];
            a.y = ldsA[lmod * 64 + ak + 1];

            // B fragment: B[k][n] = W[n][k]; N striped across lanes, K split lo/hi halves
            const int k0   = c * 64 + ak;
            const size_t brow = (size_t)(Nbase + lmod) * D_DIM + k0;
#pragma unroll
            for (int m = 0; m < 5; ++m) {
                v2f b;
                b.x = Wmat[m][brow];
                b.y = Wmat[m][brow + 1];
                acc[m] = __builtin_amdgcn_wmma_f32_16x16x4_f32(
                    /*neg_a=*/false, a, /*neg_b=*/false, b,
                    /*c_mod=*/(short)0, acc[m],
                    /*reuse_a=*/false, /*reuse_b=*/false);
            }
        }
    }

    // C/D layout: VGPR r -> M = r (lanes 0-15) or r+8 (lanes 16-31), N = lmod.
    // Stage k (acc[0]) and q (acc[2]) logits into LDS for row-softmax over N=64;
    // store v directly, sigmoid(e), sigmoid(w) in-register.
#pragma unroll
    for (int r = 0; r < 8; ++r) {
        const int m = hi ? r + 8 : r;
        const int n = Nbase + lmod;
        ldsS[m * 64 + n]        = acc[0][r];
        ldsS[1024 + m * 64 + n] = acc[2][r];
        const size_t gi = (size_t)(Mbase + m) * N_DIM + n;
        outv[gi] = acc[1][r];
        oute[gi] = 1.0f / (1.0f + __expf(-acc[3][r]));
        outw[gi] = 1.0f / (1.0f + __expf(-acc[4][r]));
    }
    __syncthreads();

    // Softmax over N=64 per row (16 rows handled by 16 threads)
    if (tid < 16) {
#pragma unroll
        for (int s = 0; s < 2; ++s) {
            float* rowp = &ldsS[s * 1024 + tid * 64];
            float mx = rowp[0];
            for (int j = 1; j < 64; ++j) mx = fmaxf(mx, rowp[j]);
            float sum = 0.0f;
            for (int j = 0; j < 64; ++j) { float e2 = __expf(rowp[j] - mx); rowp[j] = e2; sum += e2; }
            const float inv = 1.0f / sum;
            for (int j = 0; j < 64; ++j) rowp[j] *= inv;
        }
    }
    __syncthreads();

    // Cooperative coalesced store of softmaxed k and q
#pragma unroll
    for (int i = tid; i < 1024; i += 128) {
        const int m = i >> 6, n = i & 63;
        const size_t gi = (size_t)(Mbase + m) * N_DIM + n;
        outk[gi] = ldsS[i];
        outq[gi] = ldsS[1024 + i];
    }
}

// ---------------------------------------------------------------------------
// Kernel 2: sequential scan over T. One workgroup per batch (64 blocks),
// 64 threads; thread i owns state row S[i][0..63] in 64 VGPRs (fully
// unrolled so indexing stays in registers). Per-step vectors broadcast
// through LDS.
// ---------------------------------------------------------------------------
__global__ __launch_bounds__(64) void ntm_scan_kernel(
    const float* __restrict__ S0,
    const float* __restrict__ kG, const float* __restrict__ vG,
    const float* __restrict__ qG, const float* __restrict__ eG,
    const float* __restrict__ wG,
    float* __restrict__ out, float* __restrict__ Sfin)
{
    __shared__ float kb[64], vb[64], qb[64], eb[64], wb[64];
    const int b = blockIdx.x;
    const int i = threadIdx.x;

    float S[64];
#pragma unroll
    for (int j = 0; j < 64; ++j)
        S[j] = S0[(size_t)b * 4096 + (size_t)i * 64 + j];

    for (int t = 0; t < T_DIM; ++t) {
        const size_t base = (size_t)t * 4096 + (size_t)b * 64 + i;
        kb[i] = kG[base]; vb[i] = vG[base]; qb[i] = qG[base];
        eb[i] = eG[base]; wb[i] = wG[base];
        __syncthreads();

        const float ei  = eb[i];
        const float wvi = wb[i] * vb[i];
        float acc = 0.0f;
#pragma unroll
        for (int j = 0; j < 64; ++j) {
            const float kj = kb[j];
            float s = S[j];
            s = s * (1.0f - ei * kj) + wvi * kj;   // erase + write (rank-1)
            S[j] = s;
            acc = fmaf(s, qb[j], acc);             // read head dot product
        }
        out[(size_t)t * 4096 + (size_t)b * 64 + i] = tanhf(acc);
        __syncthreads();   // protect LDS buffers before next step's load
    }

#pragma unroll
    for (int j = 0; j < 64; ++j)
        Sfin[(size_t)b * 4096 + (size_t)i * 64 + j] = S[j];
}

// ---------------------------------------------------------------------------
extern "C" void kernel_launch(void* const* d_in, const int* in_sizes, int n_in,
                              void* d_out, int out_size, void* d_ws, size_t ws_size,
                              hipStream_t stream) {
    (void)in_sizes; (void)n_in; (void)out_size; (void)ws_size;
    const float* x  = (const float*)d_in[0];
    const float* S0 = (const float*)d_in[1];
    const float* Wk = (const float*)d_in[2];
    const float* Wv = (const float*)d_in[3];
    const float* Wq = (const float*)d_in[4];
    const float* We = (const float*)d_in[5];
    const float* Ww = (const float*)d_in[6];

    float* out = (float*)d_out;            // (T,B,N) output, then (B,N,N) S_final
    float* ws  = (float*)d_ws;
    float* wk  = ws;
    float* wv  = ws + TBN;
    float* wq  = ws + 2 * TBN;
    float* we_ = ws + 3 * TBN;
    float* ww_ = ws + 4 * TBN;

    ntm_proj_kernel<<<ROWS / 16, 128, 0, stream>>>(
        x, Wk, Wv, Wq, We, Ww, wk, wv, wq, we_, ww_);

    ntm_scan_kernel<<<B_DIM, 64, 0, stream>>>(
        S0, wk, wv, wq, we_, ww_, out, out + TBN);
}